// SlayerNet_8383776162005
// MI455X (gfx1250) — compile-verified
//
#include <hip/hip_runtime.h>
#include <hip/hip_bf16.h>

typedef __attribute__((ext_vector_type(2))) float v2f;
typedef __attribute__((ext_vector_type(8))) float v8f;

#define N_IN 250
#define N_HID 25
#define T_LEN 501
#define THETA 10.0f

#define KP 256   // padded K
#define NP 512   // padded N (also padded P1 row stride)

// ---------------------------------------------------------------------------
// Kernel 0: zero-pad operands so the WMMA loop is branch-free.
//   Wpad: 32 x 256  (w1, M and K padded)
//   Xpad: 256 x 512 (X,  K and N padded)
// ---------------------------------------------------------------------------
__global__ void slayer_pad(const float* __restrict__ w1,
                           const float* __restrict__ X,
                           float* __restrict__ Wpad,
                           float* __restrict__ Xpad) {
  const int idx = blockIdx.x * blockDim.x + threadIdx.x;
  if (idx < KP * NP) {
    const int kk = idx >> 9;        // / NP
    const int n  = idx & (NP - 1);
    Xpad[idx] = (kk < N_IN && n < T_LEN) ? X[kk * T_LEN + n] : 0.0f;
  }
  if (idx < 32 * KP) {
    const int m  = idx >> 8;        // / KP
    const int kk = idx & (KP - 1);
    Wpad[idx] = (m < N_HID && kk < N_IN) ? w1[m * N_IN + kk] : 0.0f;
  }
}

// ---------------------------------------------------------------------------
// Kernel A: P1p (32x512, padded) = w1 @ X via V_WMMA_F32_16X16X4_F32.
// 2 M-tiles x 32 N-tiles = 64 waves; one 16x16 f32 tile per wave.
// Padded operands AND padded output -> zero conditionals anywhere.
// ---------------------------------------------------------------------------
__global__ void slayer_gemm_wmma(const float* __restrict__ Wpad,
                                 const float* __restrict__ Xpad,
                                 float* __restrict__ P1p) {
  const int wave = (blockIdx.x * blockDim.x + threadIdx.x) >> 5;
  const int lane = threadIdx.x & 31;
  if (wave >= 64) return;

  const int mt = wave & 1;      // M tiles of 16 (covers 25 rows)
  const int nt = wave >> 1;     // N tiles of 16 (covers 501 cols)
  const int m0 = mt * 16;
  const int n0 = nt * 16;

  const int half = lane >> 4;   // selects K-pair within each K=4 step
  const int l16  = lane & 15;
  const int n    = n0 + l16;

  // A layout (16x4 MxK): lane -> M = lane%16, VGPR v -> K = 2*(lane/16)+v.
  // Per lane the A pair is contiguous in K => single b64 load.
  const float* ap = Wpad + (m0 + l16) * KP + 2 * half;
  // B layout mirrored: VGPR v -> K = 2*(lane/16)+v, lane%16 -> N.
  const float* bp = Xpad + (2 * half) * NP + n;

  v8f c = {};
#pragma unroll 4
  for (int kk = 0; kk < KP; kk += 4) {
    v2f a = *(const v2f*)ap;          // Wpad[m][ka], Wpad[m][ka+1]
    v2f b;
    b.x = bp[0];                      // Xpad[ka    ][n]
    b.y = bp[NP];                     // Xpad[ka + 1][n]
    c = __builtin_amdgcn_wmma_f32_16x16x4_f32(
        /*neg_a=*/false, a, /*neg_b=*/false, b,
        /*c_mod=*/(short)0, c, /*reuse_a=*/false, /*reuse_b=*/false);
    ap += 4;
    bp += 4 * NP;
  }

  // C/D layout: VGPR r, lanes 0-15 -> M = r, lanes 16-31 -> M = r + 8.
  // Padded destination: store all 8 rows unconditionally.
  float* dst = P1p + (m0 + 8 * half) * NP + n;
#pragma unroll
  for (int r = 0; r < 8; ++r) dst[r * NP] = c[r];
}

// ---------------------------------------------------------------------------
// Kernel B (single workgroup): causal srm-conv on P1p -> scan1 (wave 0, one
// lane per hidden neuron; sigma==1 collapses the refractory buffer to a
// shared R-vector) -> P2[t] = w2 . spikes(t) folded into the scan ->
// srm-conv -> scalar scan2 -> d_out.
// ---------------------------------------------------------------------------
__global__ void slayer_fused_tail(const float* __restrict__ P1p,
                                  const float* __restrict__ srm, int Lsrm,
                                  const float* __restrict__ ref, int R,
                                  const float* __restrict__ w2,
                                  float* __restrict__ out) {
  __shared__ float taps[512];             // causal taps, taps[i] = srm[c - i]
  __shared__ float pots1[N_HID * T_LEN];  // layer-1 potentials
  __shared__ float P2[T_LEN];             // w2 @ spikes1
  __shared__ float pots2[T_LEN];          // layer-2 potentials

  const int tid = threadIdx.x;
  const int center = Lsrm >> 1;           // srm is zero above its center
  int k = center + 1;                     // number of causal taps
  if (k > 512) k = 512;
  if (R > 64) R = 64;

  for (int i = tid; i < k; i += blockDim.x) taps[i] = srm[center - i];
  __syncthreads();

  // ---- Phase 1: pots1[n][t] = sum_i taps[i] * P1p[n][t-i] ------------------
  for (int idx = tid; idx < N_HID * T_LEN; idx += blockDim.x) {
    const int nn = idx / T_LEN;
    const int t  = idx - nn * T_LEN;
    const float* row = P1p + nn * NP;     // padded row stride
    const int imax = (t < k - 1) ? t : (k - 1);
    float s = 0.0f;
    for (int i = 0; i <= imax; ++i) s += taps[i] * row[t - i];
    pots1[idx] = s;
  }
  __syncthreads();

  // ---- Phase 2: sequential scan over T, wave 0, lane = neuron --------------
  if (tid < 32) {
    const int lane = tid;
    const bool live = (lane < N_HID);
    const float w2v = live ? w2[lane] : 0.0f;
    float aP[64];                         // shared refractory state (uniform)
    for (int j = 0; j < 64; ++j) aP[j] = 0.0f;

    for (int t = 0; t < T_LEN; ++t) {
      const float u = (live ? pots1[lane * T_LEN + t] : -1.0e30f) + aP[0];
      const bool spk = live && (u > THETA);
      float numf = spk ? 1.0f : 0.0f;     // spike count across neurons
      float dotf = spk ? w2v : 0.0f;      // w2 . spikes
      for (int off = 16; off > 0; off >>= 1) {
        numf += __shfl_xor(numf, off, 32);
        dotf += __shfl_xor(dotf, off, 32);
      }
      if (lane == 0) P2[t] = dotf;
      // a_next[j] = a[j+1] + ref[j+1]*num ; a_next[R-1] = 0
      for (int j = 0; j + 1 < R; ++j) aP[j] = aP[j + 1] + ref[j + 1] * numf;
      aP[R - 1] = 0.0f;
    }
  }
  __syncthreads();

  // ---- Phase 3: pots2[t] = sum_i taps[i] * P2[t-i] -------------------------
  for (int t = tid; t < T_LEN; t += blockDim.x) {
    const int imax = (t < k - 1) ? t : (k - 1);
    float s = 0.0f;
    for (int i = 0; i <= imax; ++i) s += taps[i] * P2[t - i];
    pots2[t] = s;
  }
  __syncthreads();

  // ---- Phase 4: scalar scan for the single output neuron -------------------
  if (tid == 0) {
    float aP[64];
    for (int j = 0; j < 64; ++j) aP[j] = 0.0f;
    for (int t = 0; t < T_LEN; ++t) {
      const float u = pots2[t] + aP[0];
      const float spk = (u > THETA) ? 1.0f : 0.0f;
      out[t] = spk;                       // spk / T_S with T_S == 1
      for (int j = 0; j + 1 < R; ++j) aP[j] = aP[j + 1] + ref[j + 1] * spk;
      aP[R - 1] = 0.0f;
    }
  }
}

extern "C" void kernel_launch(void* const* d_in, const int* in_sizes, int n_in,
                              void* d_out, int out_size, void* d_ws, size_t ws_size,
                              hipStream_t stream) {
  const float* x   = (const float*)d_in[0];   // (1,1,1,250,501) -> (250,501)
  const float* w1  = (const float*)d_in[1];   // (25,250)
  const float* w2  = (const float*)d_in[2];   // (1,25)
  const float* srm = (const float*)d_in[3];   // full symmetric-padded kernel
  const float* ref = (const float*)d_in[4];   // refractory kernel
  const int Lsrm = in_sizes[3];
  const int R    = in_sizes[4];

  // Workspace layout (floats): P1p (32x512) | Wpad (32x256) | Xpad (256x512)
  float* P1p  = (float*)d_ws;
  float* Wpad = P1p + 32 * NP;
  float* Xpad = Wpad + 32 * KP;

  // Prologue: zero-pad operands (131072 lanes cover Xpad; first 8192 do Wpad).
  slayer_pad<<<(KP * NP + 255) / 256, 256, 0, stream>>>(w1, x, Wpad, Xpad);
  // 64 waves = 8 blocks x 8 waves for the fully branch-free WMMA GEMM.
  slayer_gemm_wmma<<<8, 256, 0, stream>>>(Wpad, Xpad, P1p);
  // Single workgroup for everything sequential.
  slayer_fused_tail<<<1, 256, 0, stream>>>(P1p, srm, Lsrm, ref, R, w2,
                                           (float*)d_out);
}